// microGIF_weights_only_11347303596329
// MI455X (gfx1250) — compile-verified
//
#include <hip/hip_runtime.h>
#include <cstdint>
#include <cstddef>

#define NN 8192
#define TK 128            // K-tile (floats)
#define PITCH 132         // padded LDS row pitch (floats): bank stride 4 -> conflict-free
#define ROWS 64           // rows per workgroup (4 waves x 16)

typedef __attribute__((ext_vector_type(2))) float v2f;
typedef __attribute__((ext_vector_type(4))) float v4f;
typedef __attribute__((ext_vector_type(8))) float v8f;

__device__ __forceinline__ uint32_t rotl32(uint32_t x, uint32_t n) {
    return (x << n) | (x >> (32u - n));
}

// CDNA5 async global->LDS copy (bypasses VGPRs, tracked on ASYNCcnt)
__device__ __forceinline__ void async_load_b128(uint32_t lds_byte, const float* g) {
    asm volatile("global_load_async_to_lds_b128 %0, %1, off"
                 :: "v"(lds_byte), "v"(g)
                 : "memory");
}

// ---------------------------------------------------------------------------
// Kernel A: elementwise prep
// ---------------------------------------------------------------------------
__global__ void gif_prep(const float* __restrict__ tss,
                         const float* __restrict__ tau_s,
                         const float* __restrict__ theta_v,
                         const float* __restrict__ theta_inf,
                         const float* __restrict__ J_theta,
                         const float* __restrict__ spiked,
                         const float* __restrict__ tau_theta,
                         float* __restrict__ eps_ws,
                         float* __restrict__ theta_ws) {
    int i = blockIdx.x * blockDim.x + threadIdx.x;
    if (i >= NN) return;
    float th = theta_v[i];
    theta_ws[i] = th + (theta_inf[i] - th + J_theta[i] * spiked[i]) / tau_theta[i];
    float t  = tss[i] - 1.0f;           // DELTA_DELAY = 1.0
    float ts = tau_s[i];
    eps_ws[i] = (1.0f + tanhf(t)) * expf(-t / ts) / ts;
}

// ---------------------------------------------------------------------------
// Kernel B: I_syn = w @ eps.
//  - w streamed HBM -> LDS with global_load_async_to_lds_b128 (coalesced 512B
//    per instruction, double-buffered K-tiles, s_wait_asynccnt pipeline)
//  - V_WMMA_F32_16X16X4_F32 reduces 16 rows per wave; A fragments read from
//    the padded LDS tile (bank-conflict-free ds_load_b64), eps in column 0 of B.
// ---------------------------------------------------------------------------
__global__ __launch_bounds__(128) void gif_matvec_wmma(
        const float* __restrict__ w,
        const float* __restrict__ eps_ws,
        float* __restrict__ isyn_ws) {
    __shared__ float s_eps[NN];                    // 32 KB
    __shared__ float s_tile[2][ROWS * PITCH];      // 2 x 33 KB, double buffer

    const int tid     = threadIdx.x;
    const int lane    = tid & 31;
    const int wave    = tid >> 5;
    const int rowBase = blockIdx.x * ROWS;

    // ---- issue async loads for K-tile 0 (overlaps with the eps copy) ----
    // tile = 64 rows x 128 floats; 2048 b128 slots; 16 per thread
    {
        float* dst = &s_tile[0][0];
        #pragma unroll
        for (int it = 0; it < 16; ++it) {
            int s    = it * 128 + tid;
            int row  = s >> 5;              // 32 x b128 per 512B row chunk
            int c16  = s & 31;
            uint32_t lds = (uint32_t)(uintptr_t)(dst + row * PITCH + c16 * 4);
            const float* g = w + (size_t)(rowBase + row) * NN + c16 * 4;
            async_load_b128(lds, g);
        }
    }

    // ---- stage eps into LDS (regular coalesced b128 loads) ----
    for (int t = tid; t < NN / 4; t += blockDim.x) {
        ((v4f*)s_eps)[t] = ((const v4f*)eps_ws)[t];
    }

    const int   kl   = (lane >> 4) << 1;                   // 0 or 2
    const float colm = ((lane & 15) == 0) ? 1.0f : 0.0f;   // B column N=0 only
    const int   rowL = wave * 16 + (lane & 15);

    v8f acc = {};
    const int NT = NN / TK;                                // 64 K-tiles

    for (int kt = 0; kt < NT; ++kt) {
        if (kt + 1 < NT) {
            // prefetch next tile into the other buffer
            float* dst = &s_tile[(kt + 1) & 1][0];
            const size_t kofs = (size_t)(kt + 1) * TK;
            #pragma unroll
            for (int it = 0; it < 16; ++it) {
                int s    = it * 128 + tid;
                int row  = s >> 5;
                int c16  = s & 31;
                uint32_t lds = (uint32_t)(uintptr_t)(dst + row * PITCH + c16 * 4);
                const float* g = w + (size_t)(rowBase + row) * NN + kofs + c16 * 4;
                async_load_b128(lds, g);
            }
            // <=16 outstanding: current tile's 16 async ops have retired
            asm volatile("s_wait_asynccnt 0x10" ::: "memory");
        } else {
            asm volatile("s_wait_asynccnt 0x0" ::: "memory");
        }
        __syncthreads();

        const float* arow = &s_tile[kt & 1][0] + rowL * PITCH + kl;
        const float* ep   = &s_eps[kt * TK + kl];
        #pragma unroll 8
        for (int kk = 0; kk < TK; kk += 4) {
            v2f a = *(const v2f*)(arow + kk);     // ds_load_b64, conflict-free
            v2f e = *(const v2f*)(ep + kk);       // broadcast ds_load_b64
            v2f b;
            b.x = e.x * colm;
            b.y = e.y * colm;
            acc = __builtin_amdgcn_wmma_f32_16x16x4_f32(
                      /*neg_a=*/false, a, /*neg_b=*/false, b,
                      /*c_mod=*/(short)0, acc,
                      /*reuse_a=*/false, /*reuse_b=*/false);
        }
        __syncthreads();
    }

    // D layout: VGPR r, lane 0 -> (M=r, N=0); lane 16 -> (M=8+r, N=0)
    if ((lane & 15) == 0) {
        int mb = rowBase + wave * 16 + (lane >> 4) * 8;
        #pragma unroll
        for (int r = 0; r < 8; ++r) isyn_ws[mb + r] = acc[r];
    }
}

// ---------------------------------------------------------------------------
// Kernel C: epilogue (diagonal fixup, membrane update, threefry Bernoulli)
//   out[0:N) = spikes_lambda, out[N:2N) = spiked_new, out[2N:3N) = v_new
// ---------------------------------------------------------------------------
__global__ void gif_update(const float* __restrict__ I_ext,
                           const float* __restrict__ w,
                           const float* __restrict__ v,
                           const float* __restrict__ tss,
                           const float* __restrict__ tau_m,
                           const float* __restrict__ E_L,
                           const float* __restrict__ c,
                           const float* __restrict__ Delta_u,
                           const float* __restrict__ eps_ws,
                           const float* __restrict__ theta_ws,
                           const float* __restrict__ isyn_ws,
                           float* __restrict__ out) {
    int i = blockIdx.x * blockDim.x + threadIdx.x;
    if (i >= NN) return;

    // remove self-recurrence: I_syn[i] -= w[i][i] * eps[i]
    float isyn = isyn_ws[i] - w[(size_t)i * NN + i] * eps_ws[i];

    float vi = v[i];
    float v_next = vi + (E_L[i] - vi + I_ext[i]) / tau_m[i] + isyn;
    float nr  = (tss[i] > 2.0f) ? 1.0f : 0.0f;        // T_REFRACTORY = 2.0
    float lam = nr * c[i] * expf((v_next - theta_ws[i]) / Delta_u[i]);
    lam = fminf(fmaxf(lam, 0.0f), 1.0f);

    // threefry2x32, key = threefry_seed(42) = (0, 42); counts = iota(N)
    const uint32_t kk0 = 0u, kk1 = 42u;
    const uint32_t kk2 = kk0 ^ kk1 ^ 0x1BD11BDAu;
    const uint32_t ks[3] = {kk0, kk1, kk2};
    const uint32_t rot[8] = {13u, 15u, 26u, 6u, 17u, 29u, 16u, 24u};

    uint32_t j  = (uint32_t)(i & 4095);
    uint32_t x0 = j;                 // counts[0:4096]
    uint32_t x1 = j + 4096u;         // counts[4096:8192]
    x0 += ks[0]; x1 += ks[1];
    #pragma unroll
    for (int g = 0; g < 5; ++g) {
        const uint32_t* R = &rot[(g & 1) * 4];
        #pragma unroll
        for (int r = 0; r < 4; ++r) {
            x0 += x1;
            x1 = rotl32(x1, R[r]);
            x1 ^= x0;
        }
        x0 += ks[(g + 1) % 3];
        x1 += ks[(g + 2) % 3] + (uint32_t)(g + 1);
    }
    uint32_t bits = (i < 4096) ? x0 : x1;

    // jax.random.uniform: bitcast((bits>>9)|0x3f800000) - 1.0 ; bernoulli: u < p
    float u = __uint_as_float((bits >> 9) | 0x3f800000u) - 1.0f;
    float s = (u < lam) ? 1.0f : 0.0f;

    out[i]          = lam;
    out[NN + i]     = s;
    out[2 * NN + i] = (1.0f - s) * v_next;            // RESET_POTENTIAL = 0
}

// ---------------------------------------------------------------------------
extern "C" void kernel_launch(void* const* d_in, const int* in_sizes, int n_in,
                              void* d_out, int out_size, void* d_ws, size_t ws_size,
                              hipStream_t stream) {
    const float* I_ext    = (const float*)d_in[0];
    const float* w        = (const float*)d_in[1];
    const float* v        = (const float*)d_in[2];
    const float* spiked   = (const float*)d_in[3];
    const float* tss      = (const float*)d_in[4];
    const float* theta_v  = (const float*)d_in[5];
    const float* tau_m    = (const float*)d_in[6];
    const float* tau_s    = (const float*)d_in[7];
    const float* tau_th   = (const float*)d_in[8];
    const float* J_theta  = (const float*)d_in[9];
    const float* E_L      = (const float*)d_in[10];
    const float* c        = (const float*)d_in[11];
    const float* Delta_u  = (const float*)d_in[12];
    const float* th_inf   = (const float*)d_in[13];
    float* out = (float*)d_out;

    float* eps_ws   = (float*)d_ws;          // N floats
    float* theta_ws = eps_ws + NN;           // N floats
    float* isyn_ws  = theta_ws + NN;         // N floats

    gif_prep<<<NN / 256, 256, 0, stream>>>(tss, tau_s, theta_v, th_inf,
                                           J_theta, spiked, tau_th,
                                           eps_ws, theta_ws);

    // 128 blocks x 4 waves; each block streams 64 rows through LDS tiles
    gif_matvec_wmma<<<NN / ROWS, 128, 0, stream>>>(w, eps_ws, isyn_ws);

    gif_update<<<NN / 256, 256, 0, stream>>>(I_ext, w, v, tss, tau_m, E_L,
                                             c, Delta_u, eps_ws, theta_ws,
                                             isyn_ws, out);
}